// GAT_1ly_66709432041780
// MI455X (gfx1250) — compile-verified
//
#include <hip/hip_runtime.h>
#include <math.h>

#define N_NODES 50000
#define N_EDGES 800000
#define IN_DIM 128
#define HIDDEN 64
#define NEG_SLOPE 0.2f
#define E_TOT (N_EDGES + N_NODES)   // edges + self loops = 850000

typedef float v2f __attribute__((ext_vector_type(2)));
typedef float v8f __attribute__((ext_vector_type(8)));

// ---------------------------------------------------------------------------
// float atomic max via signed/unsigned integer atomics (works with -inf init,
// mixed-sign racing is safe due to IEEE-754 ordered bit patterns)
// ---------------------------------------------------------------------------
__device__ __forceinline__ void atomicMaxFloat(float* addr, float val) {
    if (val >= 0.0f) {
        atomicMax((int*)addr, __float_as_int(val));
    } else {
        atomicMin((unsigned int*)addr, __float_as_uint(val));
    }
}

// ---------------------------------------------------------------------------
// 0) init workspace: m = -inf, s = 0, out_acc = 0
//    grid sized so blockDim*gridDim == N_NODES*HIDDEN exactly (3,200,000)
// ---------------------------------------------------------------------------
__global__ void init_ws_kernel(float* __restrict__ mbuf, float* __restrict__ sbuf,
                               float* __restrict__ outacc) {
    int i = blockIdx.x * blockDim.x + threadIdx.x;
    if (i < N_NODES) { mbuf[i] = -INFINITY; sbuf[i] = 0.0f; }
    outacc[i] = 0.0f;
}

// ---------------------------------------------------------------------------
// 1) h = x @ W^T via V_WMMA_F32_16X16X4_F32, fused with a_src/a_dst dots.
//    One wave (32 threads) per 16-row tile; 3125 tiles exactly.
//    A (16x4 f32):  lane<16 -> {K0,K1}, lane>=16 -> {K2,K3}, M = lane&15
//    B (4x16 f32):  lane<16 -> {K0,K1}, lane>=16 -> {K2,K3}, N = lane&15
//    C/D (16x16):   vgpr r, lane l -> (M = r + 8*(l>>4), N = l&15)
// ---------------------------------------------------------------------------
__global__ void __launch_bounds__(32)
gat_gemm_wmma_kernel(const float* __restrict__ x, const float* __restrict__ W,
                     const float* __restrict__ att_src_v, const float* __restrict__ att_dst_v,
                     float* __restrict__ h, float* __restrict__ a_src,
                     float* __restrict__ a_dst)
{
    const int tile = blockIdx.x;        // 0 .. 3124
    const int lane = threadIdx.x;       // 0 .. 31 (wave32)
    const int half = lane >> 4;
    const int lrow = lane & 15;
    const int row0 = tile * 16;

    const float* xrow  = x + (size_t)(row0 + lrow) * IN_DIM;
    const float* wrow  = W + (size_t)lrow * IN_DIM;   // W is [HIDDEN, IN_DIM] row-major

    v8f acc0 = {}, acc1 = {}, acc2 = {}, acc3 = {};

    for (int kb = 0; kb < IN_DIM; kb += 4) {
        const int k = kb + 2 * half;                 // even -> 8B aligned float2 loads
        v2f a  = *(const v2f*)(xrow + k);            // A fragment (shared by 4 col tiles)
        v2f b0 = *(const v2f*)(wrow + k);                      // cols  0..15 = W rows  0..15
        v2f b1 = *(const v2f*)(wrow + 16 * IN_DIM + k);        // cols 16..31
        v2f b2 = *(const v2f*)(wrow + 32 * IN_DIM + k);        // cols 32..47
        v2f b3 = *(const v2f*)(wrow + 48 * IN_DIM + k);        // cols 48..63
        acc0 = __builtin_amdgcn_wmma_f32_16x16x4_f32(false, a, false, b0, (short)0, acc0, false, false);
        acc1 = __builtin_amdgcn_wmma_f32_16x16x4_f32(false, a, false, b1, (short)0, acc1, false, false);
        acc2 = __builtin_amdgcn_wmma_f32_16x16x4_f32(false, a, false, b2, (short)0, acc2, false, false);
        acc3 = __builtin_amdgcn_wmma_f32_16x16x4_f32(false, a, false, b3, (short)0, acc3, false, false);
    }

    const float as0 = att_src_v[lrow],      ad0 = att_dst_v[lrow];
    const float as1 = att_src_v[16 + lrow], ad1 = att_dst_v[16 + lrow];
    const float as2 = att_src_v[32 + lrow], ad2 = att_dst_v[32 + lrow];
    const float as3 = att_src_v[48 + lrow], ad3 = att_dst_v[48 + lrow];

    #pragma unroll
    for (int r = 0; r < 8; ++r) {
        const int row = row0 + r + 8 * half;
        float* hr = h + (size_t)row * HIDDEN;
        hr[lrow]      = acc0[r];
        hr[16 + lrow] = acc1[r];
        hr[32 + lrow] = acc2[r];
        hr[48 + lrow] = acc3[r];
        float ps = acc0[r] * as0 + acc1[r] * as1 + acc2[r] * as2 + acc3[r] * as3;
        float pd = acc0[r] * ad0 + acc1[r] * ad1 + acc2[r] * ad2 + acc3[r] * ad3;
        // reduce across the 16 lanes of this half (xor masks < 16 stay in-half)
        #pragma unroll
        for (int off = 1; off < 16; off <<= 1) {
            ps += __shfl_xor(ps, off, 32);
            pd += __shfl_xor(pd, off, 32);
        }
        if (lrow == 0) { a_src[row] = ps; a_dst[row] = pd; }
    }
}

// ---------------------------------------------------------------------------
// 2) per-edge leaky_relu score -> segment max (atomic)
// ---------------------------------------------------------------------------
__global__ void edge_max_kernel(const int* __restrict__ ei,
                                const float* __restrict__ a_src,
                                const float* __restrict__ a_dst,
                                float* __restrict__ mbuf) {
    int e = blockIdx.x * blockDim.x + threadIdx.x;
    if (e >= E_TOT) return;
    int s, d;
    if (e < N_EDGES) { s = ei[e]; d = ei[N_EDGES + e]; }
    else             { s = d = e - N_EDGES; }             // self loop
    float v = a_src[s] + a_dst[d];
    v = (v > 0.0f) ? v : v * NEG_SLOPE;
    atomicMaxFloat(&mbuf[d], v);
}

// ---------------------------------------------------------------------------
// 3) per-edge: ex = exp(e - m[dst]); s[dst] += ex; out[dst,:] += ex * h[src,:]
//    One wave32 per edge, 2 columns per lane (float2 gathers from L2-resident h)
// ---------------------------------------------------------------------------
__global__ void edge_scatter_kernel(const int* __restrict__ ei,
                                    const float* __restrict__ a_src,
                                    const float* __restrict__ a_dst,
                                    const float* __restrict__ mbuf,
                                    const float* __restrict__ h,
                                    float* __restrict__ sbuf,
                                    float* __restrict__ outacc) {
    int wave = (blockIdx.x * blockDim.x + threadIdx.x) >> 5;
    int lane = threadIdx.x & 31;
    if (wave >= E_TOT) return;
    int s, d;
    if (wave < N_EDGES) { s = ei[wave]; d = ei[N_EDGES + wave]; }
    else                { s = d = wave - N_EDGES; }
    float v = a_src[s] + a_dst[d];
    v = (v > 0.0f) ? v : v * NEG_SLOPE;
    float ex = expf(v - mbuf[d]);
    if (lane == 0) atomicAdd(&sbuf[d], ex);
    v2f hv = *(const v2f*)(h + (size_t)s * HIDDEN + lane * 2);
    float* ob = outacc + (size_t)d * HIDDEN + lane * 2;
    atomicAdd(ob,     ex * hv.x);
    atomicAdd(ob + 1, ex * hv.y);
}

// ---------------------------------------------------------------------------
// 4) finalize: out = relu(out_acc/s + bias); y = sigmoid(out @ W_lin^T + b_lin)
//    One wave32 per node, 2 columns per lane, full-wave butterfly reduction.
// ---------------------------------------------------------------------------
__global__ void finalize_kernel(const float* __restrict__ outacc,
                                const float* __restrict__ sbuf,
                                const float* __restrict__ bias_conv,
                                const float* __restrict__ W_lin,
                                const float* __restrict__ b_lin,
                                float* __restrict__ y) {
    int wave = (blockIdx.x * blockDim.x + threadIdx.x) >> 5;
    int lane = threadIdx.x & 31;
    if (wave >= N_NODES) return;
    float inv = 1.0f / sbuf[wave];
    v2f o = *(const v2f*)(outacc + (size_t)wave * HIDDEN + lane * 2);
    float o0 = fmaxf(o.x * inv + bias_conv[lane * 2],     0.0f);
    float o1 = fmaxf(o.y * inv + bias_conv[lane * 2 + 1], 0.0f);
    float p = o0 * W_lin[lane * 2] + o1 * W_lin[lane * 2 + 1];
    #pragma unroll
    for (int off = 1; off < 32; off <<= 1) p += __shfl_xor(p, off, 32);
    if (lane == 0) y[wave] = 1.0f / (1.0f + expf(-(p + b_lin[0])));
}

// ---------------------------------------------------------------------------
// launch
// ---------------------------------------------------------------------------
extern "C" void kernel_launch(void* const* d_in, const int* in_sizes, int n_in,
                              void* d_out, int out_size, void* d_ws, size_t ws_size,
                              hipStream_t stream) {
    const float* x         = (const float*)d_in[0];
    const int*   ei        = (const int*)  d_in[1];   // [2, N_EDGES] (int32)
    const float* W         = (const float*)d_in[2];   // [HIDDEN, IN_DIM]
    const float* att_src_v = (const float*)d_in[3];   // [HIDDEN]
    const float* att_dst_v = (const float*)d_in[4];   // [HIDDEN]
    const float* bias_conv = (const float*)d_in[5];   // [HIDDEN]
    const float* W_lin     = (const float*)d_in[6];   // [1, HIDDEN]
    const float* b_lin     = (const float*)d_in[7];   // [1]
    float* y = (float*)d_out;                         // [N_NODES]

    // workspace layout (floats): total 6,600,000 floats = 26.4 MB
    float* ws     = (float*)d_ws;
    float* h      = ws;                 // N*64 = 3,200,000
    float* a_src  = ws + 3200000;       // N
    float* a_dst  = ws + 3250000;       // N
    float* mbuf   = ws + 3300000;       // N
    float* sbuf   = ws + 3350000;       // N
    float* outacc = ws + 3400000;       // N*64 = 3,200,000

    // 0) init (grid*block == N*HIDDEN == 3,200,000 exactly)
    init_ws_kernel<<<12500, 256, 0, stream>>>(mbuf, sbuf, outacc);

    // 1) WMMA GEMM + attention dots: one wave per 16-row tile, 3125 tiles
    gat_gemm_wmma_kernel<<<3125, 32, 0, stream>>>(x, W, att_src_v, att_dst_v,
                                                  h, a_src, a_dst);

    // 2) segment max over 850,000 edges
    edge_max_kernel<<<(E_TOT + 255) / 256, 256, 0, stream>>>(ei, a_src, a_dst, mbuf);

    // 3) exp + segment sum + weighted scatter: 1 wave per edge, 8 waves/block
    edge_scatter_kernel<<<E_TOT / 8, 256, 0, stream>>>(ei, a_src, a_dst, mbuf,
                                                       h, sbuf, outacc);

    // 4) normalize + bias + relu + linear + sigmoid: 1 wave per node
    finalize_kernel<<<(N_NODES + 7) / 8, 256, 0, stream>>>(outacc, sbuf, bias_conv,
                                                           W_lin, b_lin, y);
}